// TokenResampler_43834436222995
// MI455X (gfx1250) — compile-verified
//
#include <hip/hip_runtime.h>
#include <hip/hip_bf16.h>

// ---------------------------------------------------------------------------
// TokenResampler — degenerate single-key attention:
//   softmax over a length-1 key axis == 1.0  =>  Q/Wq/bq/Wk/bk/tokens are dead.
//   out[n, s, :] = (x[n] @ Wv + bv) @ Wo + bo          (identical for all s)
//                = x[n] @ (Wv@Wo) + (bv@Wo + bo)
// Plan (MI455X roofline):
//   - 256 MB of output stores @ 23.3 TB/s  => ~11.7 us floor (HBM bound).
//   - Precompute W' = Wv@Wo (4 MB) and b' = bv@Wo+bo into d_ws once per call
//     (2.1 GFLOP, trivial) so the main pass is a SINGLE f32-WMMA GEMM.
//   - Main pass: 32 rows/block => W' L2 traffic = 128 blocks * 4 MB = 512 MB,
//     B-fragments reused across 2 row tiles (16 WMMA per 18 loads).
//   - Exact f32 path (v_wmma_f32_16x16x4_f32): store-bound kernel, no reason
//     to trade precision for matrix throughput.
// ---------------------------------------------------------------------------

typedef float v2f __attribute__((ext_vector_type(2)));
typedef float v4f __attribute__((ext_vector_type(4)));
typedef float v8f __attribute__((ext_vector_type(8)));

#define DM   1024          // d_model == k_model
#define STOK 16            // TOKEN_NUM (broadcast count)
#define LDP  (DM + 4)      // padded LDS row stride (breaks bank conflicts)

// ---------------------------------------------------------------------------
// Kernel A: W2 = Wv @ Wo   (1024x1024 f32, one 16-row tile per block)
// WMMA f32 16x16x4 operand layout (ISA 7.12.2):
//   A 16x4 : lanes 0-15 -> M=lane, v0=K+0,v1=K+1 ; lanes 16-31 -> K+2,K+3
//   B 4x16 : lanes 0-15 -> N=lane, v0=K+0,v1=K+1 ; lanes 16-31 -> K+2,K+3
// ---------------------------------------------------------------------------
__global__ __launch_bounds__(256) void wprod_kernel(
    const float* __restrict__ Wv, const float* __restrict__ Wo,
    float* __restrict__ W2) {
  __shared__ float sm[16 * LDP];

  const int tid   = threadIdx.x;
  const int m0    = blockIdx.x * 16;
  const int wave  = tid >> 5;
  const int lane  = tid & 31;
  const int lhalf = lane >> 4;
  const int lmod  = lane & 15;

  // stage 16 rows of Wv into LDS
  {
    const v4f* av = (const v4f*)(Wv + (size_t)m0 * DM);
    for (int i = tid; i < 16 * (DM / 4); i += 256) {
      const int row = i >> 8;
      const int c4  = i & 255;
      *(v4f*)&sm[row * LDP + (c4 << 2)] = av[i];
    }
  }
  __syncthreads();

  const v8f vzero = {0.f, 0.f, 0.f, 0.f, 0.f, 0.f, 0.f, 0.f};
  v8f acc[8];
#pragma unroll
  for (int t = 0; t < 8; ++t) acc[t] = vzero;

  for (int k = 0; k < DM; k += 4) {
    const v2f a = *(const v2f*)&sm[lmod * LDP + k + 2 * lhalf];
    const float* wr = Wo + (size_t)(k + 2 * lhalf) * DM + lmod;
#pragma unroll
    for (int t = 0; t < 8; ++t) {
      const int n0 = ((wave << 3) + t) << 4;
      v2f b;
      b.x = wr[n0];
      b.y = wr[n0 + DM];
      acc[t] = __builtin_amdgcn_wmma_f32_16x16x4_f32(
          false, a, false, b, (short)0, acc[t], false, false);
    }
  }

  // write W2 fragments (row = m0 + 8*lhalf + j, col = n0 + lmod)
#pragma unroll
  for (int t = 0; t < 8; ++t) {
    const int n0 = ((wave << 3) + t) << 4;
#pragma unroll
    for (int j = 0; j < 8; ++j)
      W2[(size_t)(m0 + 8 * lhalf + j) * DM + n0 + lmod] = acc[t][j];
  }
}

// ---------------------------------------------------------------------------
// Kernel B: b2 = bv @ Wo + bo   (1024-wide vector; coalesced column walk)
// ---------------------------------------------------------------------------
__global__ __launch_bounds__(256) void bias_kernel(
    const float* __restrict__ bv, const float* __restrict__ Wo,
    const float* __restrict__ bo, float* __restrict__ b2) {
  const int n = blockIdx.x * 256 + threadIdx.x;
  float acc = bo[n];
  for (int k = 0; k < DM; ++k) acc += bv[k] * Wo[(size_t)k * DM + n];
  b2[n] = acc;
}

// ---------------------------------------------------------------------------
// Kernel C: out[n, s, :] = x[n] @ W2 + b2, broadcast over s = 0..15.
// 32 rows per block (2 WMMA row tiles); each wave owns 8 column tiles and
// reuses every B fragment across both row tiles.
// ---------------------------------------------------------------------------
__global__ __launch_bounds__(256) void main_kernel(
    const float* __restrict__ x,   // (4096, 1024)
    const float* __restrict__ W2,  // (1024, 1024)
    const float* __restrict__ b2,  // (1024)
    float* __restrict__ out) {     // (4096, 16, 1024)
  __shared__ float sm[32 * LDP];   // ~128.5 KB: X tile, then Z tile

  const int tid   = threadIdx.x;
  const int m0    = blockIdx.x * 32;
  const int wave  = tid >> 5;
  const int lane  = tid & 31;
  const int lhalf = lane >> 4;
  const int lmod  = lane & 15;

  // ---- stage 32-row X tile into LDS (coalesced float4) ----
  {
    const v4f* xv = (const v4f*)(x + (size_t)m0 * DM);
    for (int i = tid; i < 32 * (DM / 4); i += 256) {
      const int row = i >> 8;
      const int c4  = i & 255;
      *(v4f*)&sm[row * LDP + (c4 << 2)] = xv[i];
    }
  }
  __syncthreads();

  const v8f vzero = {0.f, 0.f, 0.f, 0.f, 0.f, 0.f, 0.f, 0.f};
  v8f acc[2][8];
#pragma unroll
  for (int r = 0; r < 2; ++r)
#pragma unroll
    for (int t = 0; t < 8; ++t) acc[r][t] = vzero;

  // ---- Z = X @ W2 : one B fragment feeds both row tiles ----
  for (int k = 0; k < DM; k += 4) {
    const v2f a0 = *(const v2f*)&sm[lmod * LDP + k + 2 * lhalf];         // rows 0-15
    const v2f a1 = *(const v2f*)&sm[(16 + lmod) * LDP + k + 2 * lhalf];  // rows 16-31
    const float* wr = W2 + (size_t)(k + 2 * lhalf) * DM + lmod;
#pragma unroll
    for (int t = 0; t < 8; ++t) {
      const int n0 = ((wave << 3) + t) << 4;
      v2f b;
      b.x = wr[n0];
      b.y = wr[n0 + DM];
      acc[0][t] = __builtin_amdgcn_wmma_f32_16x16x4_f32(
          false, a0, false, b, (short)0, acc[0][t], false, false);
      acc[1][t] = __builtin_amdgcn_wmma_f32_16x16x4_f32(
          false, a1, false, b, (short)0, acc[1][t], false, false);
    }
  }

  // ---- Z(+bias) back into LDS ----
  __syncthreads();                       // everyone done reading X
#pragma unroll
  for (int t = 0; t < 8; ++t) {
    const int n0 = ((wave << 3) + t) << 4;
    const float bb = b2[n0 + lmod];
#pragma unroll
    for (int r = 0; r < 2; ++r)
#pragma unroll
      for (int j = 0; j < 8; ++j)        // C/D layout: row = 8*lhalf+j, col = lmod
        sm[(r * 16 + 8 * lhalf + j) * LDP + n0 + lmod] = acc[r][t][j] + bb;
  }
  __syncthreads();

  // ---- broadcast store: each Z row written STOK=16 times, contiguous b128 ----
  v4f* outv = (v4f*)(out + (size_t)m0 * STOK * DM);
  for (int i = tid; i < 32 * STOK * (DM / 4); i += 256) {
    const int row = i >> 12;             // / (STOK * DM/4) == / 4096
    const int c4  = i & 255;
    outv[i] = *(const v4f*)&sm[row * LDP + (c4 << 2)];
  }
}

extern "C" void kernel_launch(void* const* d_in, const int* in_sizes, int n_in,
                              void* d_out, int out_size, void* d_ws, size_t ws_size,
                              hipStream_t stream) {
  // setup_inputs order: x, tokens, Wq, bq, Wk, bk, Wv, bv, Wo, bo
  const float* x  = (const float*)d_in[0];
  const float* Wv = (const float*)d_in[6];
  const float* bv = (const float*)d_in[7];
  const float* Wo = (const float*)d_in[8];
  const float* bo = (const float*)d_in[9];
  float* out = (float*)d_out;

  (void)in_sizes; (void)n_in; (void)out_size; (void)ws_size;

  float* W2 = (float*)d_ws;                       // 1024*1024 f32 = 4 MB
  float* b2 = W2 + (size_t)DM * DM;               // 1024 f32

  // per-call precompute (deterministic, same stream => ordered)
  wprod_kernel<<<dim3(DM / 16), dim3(256), 0, stream>>>(Wv, Wo, W2);
  bias_kernel<<<dim3(DM / 256), dim3(256), 0, stream>>>(bv, Wo, bo, b2);

  const int N = 8 * 512;
  main_kernel<<<dim3(N / 32), dim3(256), 0, stream>>>(x, W2, b2, out);
}